// BayesianTorchModel_37022618092111
// MI455X (gfx1250) — compile-verified
//
#include <hip/hip_runtime.h>
#include <math.h>

typedef float v2f __attribute__((ext_vector_type(2)));
typedef float v8f __attribute__((ext_vector_type(8)));

#define N_NODES 512
#define KPAR 8
#define N_CONF 256
#define EPS 1e-6f
#define WAVES_PER_BLOCK 8
#define ROWS_PER_WAVE 16
#define ROWS_PER_BLOCK (WAVES_PER_BLOCK * ROWS_PER_WAVE)

#define MIN_VAL -13.815510557964274f   /* log(1e-6) */
#define MAX_VAL -1.0000005e-6f         /* log(1-1e-6) */
#define LOG2E   1.4426950408889634f
#define LN2     0.6931471805599453f

__device__ __forceinline__ float fast_exp2(float x) {
    return __builtin_amdgcn_exp2f(x);   // raw v_exp_f32 (hardware is base-2)
}

// --- Kernel 1: probability table ptab[node][conf] = sigmoid(rest_logits) ---
// Bias applied multiplicatively after exp2, so no log-domain bias add needed.
__global__ void bayes_bias_kernel(const float* __restrict__ rest_logits,
                                  float* __restrict__ ptab, int n) {
    int i = blockIdx.x * blockDim.x + threadIdx.x;
    if (i < n) {
        float x = rest_logits[i];
        ptab[i] = 1.0f / (1.0f + __expf(-x));
    }
}

// --- Kernel 2: sequential scan over nodes; each wave owns 16 batch rows ---
// Window state kept per-wave in LDS. Matrix pipeline runs in log2 domain:
//   s_diff holds (plp - log1m)*log2e  -> WMMA accumulates log2-probability sums
//   s_nMub holds the NEGATED per-row upper bound (log2 units); loaded straight
//          into the tile-invariant hoisted WMMA's C operand (no xor/negate).
//   s_S, s_l1m stay in ln units for the reference-exact clip/evidence semantics.
__global__ __launch_bounds__(256)
void bayes_scan_kernel(const float* __restrict__ evidence,
                       const float* __restrict__ root_logits,
                       const float* __restrict__ ptab,
                       float* __restrict__ out) {
    __shared__ float s_diff[WAVES_PER_BLOCK][8][16];  // (plp-log1m)*LOG2E, [slot][row]
    __shared__ float s_l1m [WAVES_PER_BLOCK][8][16];  // log1m (ln)
    __shared__ float s_S   [WAVES_PER_BLOCK][16];     // sum_k log1m (ln)
    __shared__ float s_nMub[WAVES_PER_BLOCK][16];     // -sum_k max(diff2,0) (log2)
    __shared__ float s_tmp [WAVES_PER_BLOCK][16];     // per-row weighted exp2-sums

    const int lane = threadIdx.x & 31;
    const int wave = threadIdx.x >> 5;
    const int row0 = (blockIdx.x * WAVES_PER_BLOCK + wave) * ROWS_PER_WAVE;
    const int n    = lane & 15;     // column within 16-wide tile / batch row (A)
    const int half = lane >> 4;     // 0: K slots 0,1  1: K slots 2,3

    // ---- roots: nodes 0..7 initialize the window ----
    if (lane < 16) {
        float S = 0.f, nMub = 0.f;
        #pragma unroll
        for (int j = 0; j < KPAR; ++j) {
            float x  = root_logits[j];
            float lm = fminf(x, 0.0f) - __logf(1.0f + __expf(-fabsf(x)));  // logsigmoid
            float ev = evidence[(size_t)(row0 + lane) * N_NODES + j];
            if (ev >= 0.0f) lm = __logf(fminf(fmaxf(ev, 0.f), 1.f) + EPS);
            out[(size_t)(row0 + lane) * N_NODES + j] = __expf(lm);
            float plp = fminf(fmaxf(lm, MIN_VAL), MAX_VAL);
            float y   = fminf(__expf(plp), 1.0f - EPS);
            float l1m = __logf(1.0f - y);
            float d2  = (plp - l1m) * LOG2E;
            s_diff[wave][j][lane] = d2;
            s_l1m [wave][j][lane] = l1m;
            S    += l1m;
            nMub -= fmaxf(d2, 0.f);
        }
        s_S   [wave][lane] = S;
        s_nMub[wave][lane] = nMub;
    }
    __syncthreads();

    // B-fragment for GEMM K-slots 4..7 (low 4 bits of conf = column n): tile-invariant
    v2f bf1;
    bf1.x = (float)((n >> (3 - 2 * half)) & 1);
    bf1.y = (float)((n >> (2 - 2 * half)) & 1);

    for (int node = KPAR; node < N_NODES; ++node) {
        const int rr = node - KPAR;
        const float* p_row = ptab + (size_t)rr * N_CONF;

        // C operand of the hoisted WMMA: -Mub, loaded directly (8 consecutive floats)
        v8f cshift;
        #pragma unroll
        for (int v = 0; v < 8; ++v) cshift[v] = s_nMub[wave][half * 8 + v];

        // A fragments: window diff2 values. Logical k -> physical slot (node+k)&7.
        const int kb0 = 2 * half;        // GEMM K slots for fragment 0 (window k 0..3)
        const int kb1 = 4 + 2 * half;    // fragment 1 (window k 4..7)
        v2f a0, a1;
        a0.x = s_diff[wave][(node + kb0)     & 7][n];
        a0.y = s_diff[wave][(node + kb0 + 1) & 7][n];
        a1.x = s_diff[wave][(node + kb1)     & 7][n];
        a1.y = s_diff[wave][(node + kb1 + 1) & 7][n];

        // Prefetch the whole CPT row (one clause of 16 coalesced loads) and the
        // evidence value needed by finalize, so latency hides under WMMA/exp work.
        float pbt[16];
        #pragma unroll
        for (int t = 0; t < 16; ++t) pbt[t] = p_row[t * 16 + n];
        float ev = -1.0f;
        if (lane < 16) ev = evidence[(size_t)(row0 + lane) * N_NODES + node];

        // Tile-invariant half of the dot product + shift (1 WMMA per node)
        v8f c1 = __builtin_amdgcn_wmma_f32_16x16x4_f32(false, a1, false, bf1,
                                                       (short)0, cshift, false, false);

        float ssum[8];
        #pragma unroll
        for (int v = 0; v < 8; ++v) ssum[v] = 0.f;

        #pragma unroll
        for (int t = 0; t < 16; ++t) {
            // conf = 16*t + n ; high 4 conf bits (window k 0..3) come from t
            v2f b0;
            b0.x = (float)((t >> (3 - 2 * half)) & 1);
            b0.y = (float)((t >> (2 - 2 * half)) & 1);
            v8f c = __builtin_amdgcn_wmma_f32_16x16x4_f32(false, a0, false, b0,
                                                          (short)0, c1, false, false);
            // Stage all 8 exps first (independent -> trans unit pipelines, no nops),
            // then fold in the CPT probability with fmac.
            float e[8];
            #pragma unroll
            for (int v = 0; v < 8; ++v) e[v] = fast_exp2(c[v]);
            const float pb = pbt[t];
            #pragma unroll
            for (int v = 0; v < 8; ++v) ssum[v] = fmaf(pb, e[v], ssum[v]);
        }

        // Reduce across the 16 lanes of each half (rows live in fixed VGPR index)
        #pragma unroll
        for (int v = 0; v < 8; ++v) {
            float s = ssum[v];
            s += __shfl_xor(s, 1, 32);
            s += __shfl_xor(s, 2, 32);
            s += __shfl_xor(s, 4, 32);
            s += __shfl_xor(s, 8, 32);
            ssum[v] = s;
        }
        if (lane == 0 || lane == 16) {
            #pragma unroll
            for (int v = 0; v < 8; ++v) s_tmp[wave][half * 8 + v] = ssum[v];
        }
        __syncthreads();

        // Finalize: lane l owns batch row l
        if (lane < 16) {
            float sum  = s_tmp[wave][lane];
            // back to ln domain: logm = S + (log2(sum) - nMub) * ln2
            float logm = s_S[wave][lane] + (__log2f(sum) - s_nMub[wave][lane]) * LN2;
            if (ev >= 0.0f) logm = __logf(fminf(fmaxf(ev, 0.f), 1.f) + EPS);
            out[(size_t)(row0 + lane) * N_NODES + node] = __expf(logm);

            float plp = fminf(fmaxf(logm, MIN_VAL), MAX_VAL);
            float y   = fminf(__expf(plp), 1.0f - EPS);
            float l1m = __logf(1.0f - y);
            float d2  = (plp - l1m) * LOG2E;
            const int sl = node & 7;  // evicted slot == slot of the new entry
            s_S   [wave][lane] += l1m - s_l1m[wave][sl][lane];
            s_nMub[wave][lane] += fmaxf(s_diff[wave][sl][lane], 0.f) - fmaxf(d2, 0.f);
            s_l1m [wave][sl][lane] = l1m;
            s_diff[wave][sl][lane] = d2;
        }
        __syncthreads();
    }
}

extern "C" void kernel_launch(void* const* d_in, const int* in_sizes, int n_in,
                              void* d_out, int out_size, void* d_ws, size_t ws_size,
                              hipStream_t stream) {
    const float* evidence    = (const float*)d_in[0];
    const float* root_logits = (const float*)d_in[1];
    const float* rest_logits = (const float*)d_in[2];
    float* out  = (float*)d_out;
    float* ptab = (float*)d_ws;                    // (N_NODES-K)*N_CONF floats

    const int n_bias = in_sizes[2];                // 504*256
    const int Bsz    = in_sizes[0] / N_NODES;      // 16384

    bayes_bias_kernel<<<(n_bias + 255) / 256, 256, 0, stream>>>(rest_logits, ptab, n_bias);

    const int blocks = Bsz / ROWS_PER_BLOCK;       // 128
    bayes_scan_kernel<<<blocks, WAVES_PER_BLOCK * 32, 0, stream>>>(
        evidence, root_logits, ptab, out);
}